// BiologicalMemory_61151744360675
// MI455X (gfx1250) — compile-verified
//
#include <hip/hip_runtime.h>
#include <math.h>

#define DIM  4096
#define CAP  8192
#define TOPK 8

typedef __attribute__((ext_vector_type(2))) float v2f;
typedef __attribute__((ext_vector_type(8))) float v8f;

#define NEG_BIG (-3.4028234663852886e38f)

// -----------------------------------------------------------------------------
// K1: fused (memory_bank @ query) + per-row sum-of-squares.
// One wave per 16 rows. A = 16x4 fp32 tile of the bank, B = query chunk
// replicated across all 16 columns -> D column 0 = the 16 dot products.
// fp32 A-layout: lane L (0..15): row M=L, K={0,1}; lane L+16: row M=L, K={2,3}.
// -----------------------------------------------------------------------------
__global__ __launch_bounds__(32)
void sim_rows_wmma(const float* __restrict__ mem, const float* __restrict__ q,
                   float* __restrict__ dots, float* __restrict__ mn2) {
    const int  lane = threadIdx.x;        // 0..31
    const int  half = lane >> 4;          // 0 or 1
    const int  m    = lane & 15;          // row within tile
    const long r0   = (long)blockIdx.x * 16;
    const float* rowp = mem + (r0 + m) * (long)DIM;

    v8f   acc = {};
    float ss  = 0.0f;
    for (int k0 = 0; k0 < DIM; k0 += 4) {
        const int kk = k0 + 2 * half;
        v2f a = *(const v2f*)(rowp + kk);   // 8B aligned: kk is even
        v2f b = *(const v2f*)(q + kk);      // vector chunk, same for all columns
        acc = __builtin_amdgcn_wmma_f32_16x16x4_f32(
                  false, a, false, b, (short)0, acc, false, false);
        ss = fmaf(a.x, a.x, ss);
        ss = fmaf(a.y, a.y, ss);
    }

    // Row sum-of-squares: lane m covers K%4 in {0,1}, lane m+16 covers {2,3}.
    ss += __shfl_xor(ss, 16, 32);
    if (lane < 16) mn2[r0 + m] = ss;

    // Column N=0 of D: lane 0 holds M=0..7 in acc[0..7], lane 16 holds M=8..15.
    if (lane == 0) {
        #pragma unroll
        for (int r = 0; r < 8; ++r) dots[r0 + r] = acc[r];
    } else if (lane == 16) {
        #pragma unroll
        for (int r = 0; r < 8; ++r) dots[r0 + 8 + r] = acc[r];
    }
}

// -----------------------------------------------------------------------------
// K2: query norm, weighted scores, top-8 (first-occurrence tie-break).
// Scores live in LDS (32KB of 320KB WGP LDS); 8 argmax passes.
// -----------------------------------------------------------------------------
__global__ __launch_bounds__(256)
void weight_topk(const float* __restrict__ q, const float* __restrict__ imp,
                 const float* __restrict__ age, const float* __restrict__ dots,
                 const float* __restrict__ mn2, int* __restrict__ topk) {
    __shared__ float w[CAP];
    __shared__ float rv[256];
    __shared__ int   ri[256];
    __shared__ float qn_s;
    const int tid = threadIdx.x;

    // ||query||
    float p = 0.0f;
    for (int i = tid; i < DIM; i += 256) { float v = q[i]; p = fmaf(v, v, p); }
    rv[tid] = p; __syncthreads();
    for (int s = 128; s > 0; s >>= 1) {
        if (tid < s) rv[tid] += rv[tid + s];
        __syncthreads();
    }
    if (tid == 0) qn_s = sqrtf(rv[0]);
    __syncthreads();
    const float qn = qn_s;

    // weighted = dot / max(qn*||row||, eps) * importance * exp(-0.001*age)
    for (int i = tid; i < CAP; i += 256) {
        float denom = fmaxf(qn * sqrtf(mn2[i]), 1e-8f);
        w[i] = (dots[i] / denom) * imp[i] * expf(-0.001f * age[i]);
    }
    __syncthreads();

    for (int pass = 0; pass < TOPK; ++pass) {
        float best = NEG_BIG; int bidx = 0x7fffffff;
        for (int i = tid; i < CAP; i += 256) {
            float v = w[i];
            if (v > best || (v == best && i < bidx)) { best = v; bidx = i; }
        }
        rv[tid] = best; ri[tid] = bidx;
        __syncthreads();
        for (int s = 128; s > 0; s >>= 1) {
            if (tid < s) {
                if (rv[tid + s] > rv[tid] ||
                    (rv[tid + s] == rv[tid] && ri[tid + s] < ri[tid])) {
                    rv[tid] = rv[tid + s]; ri[tid] = ri[tid + s];
                }
            }
            __syncthreads();
        }
        if (tid == 0) { topk[pass] = ri[0]; w[ri[0]] = NEG_BIG; }
        __syncthreads();
    }
}

// -----------------------------------------------------------------------------
// K3: mean-pool the 8 retrieved rows.
// -----------------------------------------------------------------------------
__global__ __launch_bounds__(256)
void gather_mean(const float* __restrict__ mem, const int* __restrict__ topk,
                 float* __restrict__ retrieved) {
    const int j = blockIdx.x * 256 + threadIdx.x;
    float s = 0.0f;
    #pragma unroll
    for (int t = 0; t < TOPK; ++t) s += mem[(long)topk[t] * DIM + j];
    retrieved[j] = s * (1.0f / TOPK);
}

// -----------------------------------------------------------------------------
// K4: out = retrieved @ W_dec.T + b_dec, same broadcast-WMMA matvec pattern.
// out[i] = sum_j retrieved[j] * W_dec[i*DIM + j] + b_dec[i]
// -----------------------------------------------------------------------------
__global__ __launch_bounds__(32)
void decode_wmma(const float* __restrict__ W, const float* __restrict__ bdec,
                 const float* __restrict__ r, float* __restrict__ out) {
    const int  lane = threadIdx.x;
    const int  half = lane >> 4;
    const int  m    = lane & 15;
    const long i0   = (long)blockIdx.x * 16;
    const float* rowp = W + (i0 + m) * (long)DIM;

    v8f acc = {};
    for (int k0 = 0; k0 < DIM; k0 += 4) {
        const int kk = k0 + 2 * half;
        v2f a = *(const v2f*)(rowp + kk);
        v2f b = *(const v2f*)(r + kk);
        acc = __builtin_amdgcn_wmma_f32_16x16x4_f32(
                  false, a, false, b, (short)0, acc, false, false);
    }

    if (lane == 0) {
        #pragma unroll
        for (int rr = 0; rr < 8; ++rr) out[i0 + rr] = acc[rr] + bdec[i0 + rr];
    } else if (lane == 16) {
        #pragma unroll
        for (int rr = 0; rr < 8; ++rr) out[i0 + 8 + rr] = acc[rr] + bdec[i0 + 8 + rr];
    }
}

// -----------------------------------------------------------------------------
// Workspace layout (floats):
//   [0,      8192)   dots
//   [8192,  16384)   mn2 (row sum-of-squares)
//   [16384, 16400)   top-k indices (int), padded to 16 for alignment
//   [16400, 20496)   retrieved (4096)
// Total: 81,984 bytes.
// -----------------------------------------------------------------------------
extern "C" void kernel_launch(void* const* d_in, const int* in_sizes, int n_in,
                              void* d_out, int out_size, void* d_ws, size_t ws_size,
                              hipStream_t stream) {
    (void)in_sizes; (void)n_in; (void)out_size; (void)ws_size;

    const float* q    = (const float*)d_in[0];   // [4096]
    const float* mem  = (const float*)d_in[1];   // [8192, 4096]
    const float* imp  = (const float*)d_in[2];   // [8192]
    const float* age  = (const float*)d_in[3];   // [8192]
    const float* W    = (const float*)d_in[4];   // [4096, 4096]
    const float* bdec = (const float*)d_in[5];   // [4096]
    float*       out  = (float*)d_out;           // [4096]

    float* dots      = (float*)d_ws;
    float* mn2       = dots + CAP;
    int*   topk      = (int*)(mn2 + CAP);
    float* retrieved = (float*)(topk + 16);

    sim_rows_wmma<<<CAP / 16, 32, 0, stream>>>(mem, q, dots, mn2);
    weight_topk  <<<1,        256, 0, stream>>>(q, imp, age, dots, mn2, topk);
    gather_mean  <<<DIM / 256, 256, 0, stream>>>(mem, topk, retrieved);
    decode_wmma  <<<DIM / 16,  32, 0, stream>>>(W, bdec, retrieved, out);
}